// CustomAttentionLayer_14851996910072
// MI455X (gfx1250) — compile-verified
//
#include <hip/hip_runtime.h>
#include <hip/hip_bf16.h>

typedef float v2f __attribute__((ext_vector_type(2)));
typedef float v8f __attribute__((ext_vector_type(8)));

#define B_DIM 64
#define T_DIM 4096
#define D_DIM 512
#define K_TOP 409           // max(1, int(4096 * 0.1))
#define NCHUNK 8            // T chunks in pass 3
#define TCHUNK (T_DIM / NCHUNK)

// ---------------------------------------------------------------------------
// Pass 1: e[b*T+t] = tanh(dot(x[b,t,:], W) + bias) via V_WMMA_F32_16X16X4_F32.
// One wave owns a 16-row tile; K-loop over D in steps of 4 -> 128 wmma / tile.
// Exact fp32 arithmetic (no precision loss vs scalar FMA).
// ---------------------------------------------------------------------------
__global__ __launch_bounds__(256) void attn_pass1_logits(
    const float* __restrict__ x, const float* __restrict__ W,
    const float* __restrict__ bias, float* __restrict__ e)
{
    __shared__ float sW[D_DIM];
    const int tid = threadIdx.x;
    for (int i = tid; i < D_DIM; i += 256) sW[i] = W[i];
    __syncthreads();

    const int wave    = tid >> 5;        // 0..7
    const int lane    = tid & 31;
    const int halfsel = lane >> 4;       // 0: K=0,1   1: K=2,3
    const int m       = lane & 15;       // A-matrix row within tile

    const long tile = (long)blockIdx.x * 8 + wave;   // 16-row tile index
    const long row0 = tile * 16;                     // global row in [0, B*T)

    const float* rowp = x + (row0 + m) * (long)D_DIM + halfsel * 2;

    v8f c = {0.f, 0.f, 0.f, 0.f, 0.f, 0.f, 0.f, 0.f};
    for (int d0 = 0; d0 < D_DIM; d0 += 4) {
        v2f a  = *(const v2f*)(rowp + d0);            // A[m][d0+2h], A[m][d0+1+2h]
        v2f bf = *(const v2f*)(&sW[d0 + halfsel * 2]); // B[k][*] = W[d0+k]
        c = __builtin_amdgcn_wmma_f32_16x16x4_f32(
                /*neg_a=*/false, a, /*neg_b=*/false, bf,
                /*c_mod=*/(short)0, c, /*reuse_a=*/false, /*reuse_b=*/false);
    }

    const float bv = bias[0];
    if (m == 0) {   // lane 0 -> rows row0..row0+7, lane 16 -> rows row0+8..+15
        #pragma unroll
        for (int g = 0; g < 8; ++g) {
            long r = row0 + g + halfsel * 8;
            e[r] = tanhf(c[g] + bv);
        }
    }
}

// ---------------------------------------------------------------------------
// Pass 2: per-batch softmax stats + exact 409th-largest threshold via 4-pass
// MSB-first radix select on order-transformed float keys. Overwrites e with
// final emphasized softmax weights a'.
// ---------------------------------------------------------------------------
__device__ __forceinline__ unsigned orderKey(float f) {
    unsigned u = __float_as_uint(f);
    return (u & 0x80000000u) ? ~u : (u | 0x80000000u);  // monotone float->uint
}

__global__ __launch_bounds__(256) void attn_pass2_weights(float* __restrict__ ew)
{
    const int b   = blockIdx.x;
    float* e      = ew + (long)b * T_DIM;
    const int tid = threadIdx.x;

    __shared__ float    red[256];
    __shared__ unsigned hist[256];
    __shared__ float    s_max, s_den;
    __shared__ unsigned s_prefix;
    __shared__ int      s_rem;

    // ---- max over T ----
    float mx = -INFINITY;
    for (int t = tid; t < T_DIM; t += 256) mx = fmaxf(mx, e[t]);
    red[tid] = mx; __syncthreads();
    for (int s = 128; s > 0; s >>= 1) {
        if (tid < s) red[tid] = fmaxf(red[tid], red[tid + s]);
        __syncthreads();
    }
    if (tid == 0) s_max = red[0];
    __syncthreads();
    const float m = s_max;

    // ---- denom ----
    float sum = 0.f;
    for (int t = tid; t < T_DIM; t += 256) sum += expf(e[t] - m);
    red[tid] = sum; __syncthreads();
    for (int s = 128; s > 0; s >>= 1) {
        if (tid < s) red[tid] += red[tid + s];
        __syncthreads();
    }
    if (tid == 0) { s_den = red[0]; s_prefix = 0u; s_rem = K_TOP; }
    __syncthreads();
    const float den = s_den;

    // ---- radix select: key of k-th largest e ----
    for (int pass = 0; pass < 4; ++pass) {
        const int shift = 24 - pass * 8;
        hist[tid] = 0u;
        __syncthreads();
        const unsigned pfx    = s_prefix;
        const unsigned himask = (pass == 0) ? 0u : ~((1u << (shift + 8)) - 1u);
        for (int t = tid; t < T_DIM; t += 256) {
            unsigned key = orderKey(e[t]);
            if ((key & himask) == pfx)
                atomicAdd(&hist[(key >> shift) & 0xFF], 1u);
        }
        __syncthreads();
        if (tid == 0) {
            int rem = s_rem;
            unsigned cum = 0;
            for (int bin = 255; bin >= 0; --bin) {
                unsigned h = hist[bin];
                if (cum + h >= (unsigned)rem) {
                    s_rem    = rem - (int)cum;
                    s_prefix = pfx | ((unsigned)bin << shift);
                    break;
                }
                cum += h;
            }
        }
        __syncthreads();
    }
    const unsigned tk = s_prefix;

    // ---- overwrite e with emphasized softmax weight ----
    for (int t = tid; t < T_DIM; t += 256) {
        float ev = e[t];
        float a  = expf(ev - m) / den;
        if (orderKey(ev) >= tk) a *= 1.5f;
        e[t] = a;
    }
}

// ---------------------------------------------------------------------------
// Pass 3: partial weighted sums over a T-chunk. Coalesced x reads (one d per
// thread), weights staged in LDS. Deterministic fixed-order accumulation.
// ---------------------------------------------------------------------------
__global__ __launch_bounds__(512) void attn_pass3_wsum(
    const float* __restrict__ x, const float* __restrict__ w,
    float* __restrict__ partial)
{
    const int ch = blockIdx.x;   // 0..NCHUNK-1
    const int b  = blockIdx.y;   // 0..B-1
    const int d  = threadIdx.x;  // 0..511

    __shared__ float sw[TCHUNK];
    const int t0 = ch * TCHUNK;
    sw[d] = w[(long)b * T_DIM + t0 + d];   // 512 threads, TCHUNK==512
    __syncthreads();

    const float* xp = x + ((long)b * T_DIM + t0) * D_DIM + d;
    float acc = 0.f;
    #pragma unroll 8
    for (int i = 0; i < TCHUNK; ++i)
        acc = fmaf(xp[(long)i * D_DIM], sw[i], acc);

    partial[((long)ch * B_DIM + b) * D_DIM + d] = acc;
}

// ---------------------------------------------------------------------------
// Pass 4: reduce NCHUNK partials into out [B*D].
// ---------------------------------------------------------------------------
__global__ __launch_bounds__(256) void attn_pass4_reduce(
    const float* __restrict__ partial, float* __restrict__ out, int n)
{
    int i = blockIdx.x * blockDim.x + threadIdx.x;
    if (i < n) {
        float s = 0.f;
        #pragma unroll
        for (int c = 0; c < NCHUNK; ++c) s += partial[(long)c * n + i];
        out[i] = s;
    }
}

extern "C" void kernel_launch(void* const* d_in, const int* in_sizes, int n_in,
                              void* d_out, int out_size, void* d_ws, size_t ws_size,
                              hipStream_t stream)
{
    const float* x    = (const float*)d_in[0];   // [64,4096,512] fp32
    const float* W    = (const float*)d_in[1];   // [512,1]       fp32
    const float* bias = (const float*)d_in[2];   // [1]           fp32
    float*       out  = (float*)d_out;           // [64,1,512]    fp32

    float* ew      = (float*)d_ws;                      // B*T   floats (e, then a')
    float* partial = ew + (size_t)B_DIM * T_DIM;        // NCHUNK*B*D floats

    // Pass 1: 16-row tiles, 8 waves/block -> (B*T/16)/8 blocks
    const int tiles  = (B_DIM * T_DIM) / 16;
    attn_pass1_logits<<<tiles / 8, 256, 0, stream>>>(x, W, bias, ew);

    // Pass 2: one block per batch
    attn_pass2_weights<<<B_DIM, 256, 0, stream>>>(ew);

    // Pass 3: partial weighted sums
    dim3 g3(NCHUNK, B_DIM);
    attn_pass3_wsum<<<g3, 512, 0, stream>>>(x, ew, partial);

    // Pass 4: final reduce
    const int n = B_DIM * D_DIM;
    attn_pass4_reduce<<<(n + 255) / 256, 256, 0, stream>>>(partial, out, n);
}